// GCCN_21388937134842
// MI455X (gfx1250) — compile-verified
//
#include <hip/hip_runtime.h>
#include <hip/hip_bf16.h>
#include <math.h>

// ---------------------------------------------------------------------------
// GCN autoencoder for MI455X (gfx1250, wave32).
//  - GEMMs: V_WMMA_F32_16X16X4_F32, templated on (CI,CO) so the K-loop fully
//    unrolls into straight-line WMMAs with no guards/branches in the hot path.
//    Out-of-range rows/cols use clamped addresses (their outputs are never
//    stored); only genuine K-tail slots are zeroed via selects.
//  - Aggregation: one wave32 per edge, float4 (b128) gathers of h[src] and
//    scalar global_atomic_add_f32 scatters. All h matrices <= 64MB -> L2.
//  - Self-loop term initializes the agg buffer (no atomics for it).
// ---------------------------------------------------------------------------

typedef __attribute__((ext_vector_type(2))) float v2f;
typedef __attribute__((ext_vector_type(8))) float v8f;

#define TPB 256
#define WPB 8   // wave32 waves per block

__global__ void k_deg_init(float* deg, int n) {
  int i = blockIdx.x * blockDim.x + threadIdx.x;
  if (i < n) deg[i] = 1.0f;  // self-loop contributes 1 to every node's degree
}

__global__ void k_deg_edges(const int* __restrict__ dst, float* deg, int E) {
  int e = blockIdx.x * blockDim.x + threadIdx.x;
  if (e < E) atomicAdd(&deg[dst[e]], 1.0f);
}

__global__ void k_rsqrt(const float* __restrict__ deg, float* dis, int n) {
  int i = blockIdx.x * blockDim.x + threadIdx.x;
  if (i < n) {
    float d = deg[i];
    dis[i] = (d > 0.0f) ? rsqrtf(d) : 0.0f;
  }
}

// Out[n,CO] = act(X[n,CI] @ W[CI,CO] (+ bias)); one wave per 16x16 tile.
// f32 WMMA layouts (ISA 7.12.2):
//   A 16x4 : lanes 0-15 row=l K={k0,k0+1}; lanes 16-31 row=l K={k0+2,k0+3}
//   B 4x16 : mirrored with columns instead of rows
//   C/D    : lane -> column, VGPR v -> row v (upper lanes: row v+8)
template <int CI, int CO>
__global__ void k_gemm_wmma(const float* __restrict__ X, const float* __restrict__ W,
                            const float* __restrict__ bias, float* __restrict__ Out,
                            int n, int total_tiles, int act) {
  constexpr int CO_T = (CO + 15) / 16;
  int tile = blockIdx.x * (blockDim.x >> 5) + (threadIdx.x >> 5);
  if (tile >= total_tiles) return;          // wave-uniform: EXEC all-ones below
  const int lane = threadIdx.x & 31;
  const int half = lane >> 4;
  const int l    = lane & 15;
  const int mt = tile / CO_T, nt = tile % CO_T;
  const int row0 = mt * 16, col0 = nt * 16;

  int arow = row0 + l;
  if (arow > n - 1) arow = n - 1;           // clamp: rows >= n never stored
  const int  bcol  = col0 + l;
  const bool colok = (CO % 16 == 0) ? true : (bcol < CO);
  const int  bcolc = colok ? bcol : (CO - 1);  // clamp: cols >= CO never stored

  const float* __restrict__ Xr = X + (size_t)arow * CI;
  const float* __restrict__ Wc = W + bcolc;

  v8f acc = {};
#pragma unroll
  for (int k0 = 0; k0 < CI; k0 += 4) {
    const int ka = k0 + half * 2;
    v2f a, b;
    if (k0 + 4 <= CI) {                     // full quad (folds at compile time)
      if (CI % 2 == 0) {
        a = *(const v2f*)(Xr + ka);         // 8B-aligned b64 load
      } else {
        a.x = Xr[ka];
        a.y = Xr[ka + 1];
      }
      b.x = Wc[(size_t)ka * CO];
      b.y = Wc[(size_t)(ka + 1) * CO];
    } else {                                // K tail: zero out-of-range K slots
      const int ka0 = (ka     < CI) ? ka     : (CI - 1);
      const int ka1 = (ka + 1 < CI) ? ka + 1 : (CI - 1);
      float ax = Xr[ka0], ay = Xr[ka1];
      float bx = Wc[(size_t)ka0 * CO], by = Wc[(size_t)ka1 * CO];
      a.x = (ka     < CI) ? ax : 0.0f;
      a.y = (ka + 1 < CI) ? ay : 0.0f;
      b.x = (ka     < CI) ? bx : 0.0f;
      b.y = (ka + 1 < CI) ? by : 0.0f;
    }
    acc = __builtin_amdgcn_wmma_f32_16x16x4_f32(
        /*neg_a=*/false, a, /*neg_b=*/false, b,
        /*c_mod=*/(short)0, acc, /*reuse_a=*/false, /*reuse_b=*/false);
  }

  if (colok) {
    const float bv = bias ? bias[bcolc] : 0.0f;
    float* __restrict__ Oc = Out + bcol;
    if (row0 + 16 <= n) {                   // common case: no row guards
#pragma unroll
      for (int v = 0; v < 8; ++v) {
        float val = acc[v] + bv;
        if (act == 1)      val = fmaxf(val, 0.0f);
        else if (act == 2) val = tanhf(val);
        Oc[(size_t)(row0 + v + half * 8) * CO] = val;
      }
    } else {
#pragma unroll
      for (int v = 0; v < 8; ++v) {
        int row = row0 + v + half * 8;
        if (row < n) {
          float val = acc[v] + bv;
          if (act == 1)      val = fmaxf(val, 0.0f);
          else if (act == 2) val = tanhf(val);
          Oc[(size_t)row * CO] = val;
        }
      }
    }
  }
}

// agg[i,:] = h[i,:] * dis[i]^2   (self-loop term; initializes agg)
__global__ void k_selfloop(const float* __restrict__ H, const float* __restrict__ dis,
                           float* __restrict__ agg, int n, int co) {
  int node = blockIdx.x * (blockDim.x >> 5) + (threadIdx.x >> 5);
  if (node >= n) return;
  int lane = threadIdx.x & 31;
  float dd = dis[node];
  dd *= dd;
  size_t base = (size_t)node * co;
  if ((co & 3) == 0) {
    const float4* __restrict__ h4 = (const float4*)(H + base);
    float4* __restrict__ a4 = (float4*)(agg + base);
    for (int f = lane; f < (co >> 2); f += 32) {
      float4 h = h4[f];
      a4[f] = make_float4(h.x * dd, h.y * dd, h.z * dd, h.w * dd);
    }
  } else {
    for (int f = lane; f < co; f += 32)
      agg[base + f] = H[base + f] * dd;
  }
}

// agg[dst,:] += h[src,:] * dis[src]*dis[dst]   (one wave32 per edge)
__global__ void k_edge_agg(const float* __restrict__ H, const int* __restrict__ src,
                           const int* __restrict__ dst, const float* __restrict__ dis,
                           float* __restrict__ agg, int E, int co) {
  int e = blockIdx.x * (blockDim.x >> 5) + (threadIdx.x >> 5);
  if (e >= E) return;
  int lane = threadIdx.x & 31;
  int s = src[e], d = dst[e];
  float nrm = dis[s] * dis[d];
  size_t hb = (size_t)s * co, ab = (size_t)d * co;
  if ((co & 3) == 0) {
    const float4* __restrict__ h4 = (const float4*)(H + hb);
    for (int f = lane; f < (co >> 2); f += 32) {
      float4 h = h4[f];
      float* a = agg + ab + ((size_t)f << 2);
      atomicAdd(a + 0, h.x * nrm);
      atomicAdd(a + 1, h.y * nrm);
      atomicAdd(a + 2, h.z * nrm);
      atomicAdd(a + 3, h.w * nrm);
    }
  } else {
    for (int f = lane; f < co; f += 32)
      atomicAdd(&agg[ab + f], H[hb + f] * nrm);
  }
}

// Out[i,:] = act(A[i,:] + bias)   (act: 0=none, 1=relu, 2=tanh)
__global__ void k_bias_act(const float* __restrict__ A, const float* __restrict__ bias,
                           float* __restrict__ Out, int n, int co, int act) {
  int node = blockIdx.x * (blockDim.x >> 5) + (threadIdx.x >> 5);
  if (node >= n) return;
  int lane = threadIdx.x & 31;
  size_t base = (size_t)node * co;
  if ((co & 3) == 0) {
    const float4* __restrict__ a4 = (const float4*)(A + base);
    float4* __restrict__ o4 = (float4*)(Out + base);
    const float4* __restrict__ b4 = (const float4*)bias;
    for (int f = lane; f < (co >> 2); f += 32) {
      float4 v = a4[f];
      float4 bb = b4[f];
      v.x += bb.x; v.y += bb.y; v.z += bb.z; v.w += bb.w;
      if (act == 1) {
        v.x = fmaxf(v.x, 0.f); v.y = fmaxf(v.y, 0.f);
        v.z = fmaxf(v.z, 0.f); v.w = fmaxf(v.w, 0.f);
      } else if (act == 2) {
        v.x = tanhf(v.x); v.y = tanhf(v.y);
        v.z = tanhf(v.z); v.w = tanhf(v.w);
      }
      o4[f] = v;
    }
  } else {
    for (int f = lane; f < co; f += 32) {
      float v = A[base + f] + bias[f];
      if (act == 1)      v = fmaxf(v, 0.0f);
      else if (act == 2) v = tanhf(v);
      Out[base + f] = v;
    }
  }
}

template <int CI, int CO>
static inline void gemm_t(const float* X, const float* W, const float* B, float* Out,
                          int n, int act, hipStream_t s) {
  constexpr int CO_T = (CO + 15) / 16;
  int total  = ((n + 15) / 16) * CO_T;
  int blocks = (total + WPB - 1) / WPB;
  k_gemm_wmma<CI, CO><<<blocks, TPB, 0, s>>>(X, W, B, Out, n, total, act);
}

// One GCN layer: H = X@W ; agg = selfloop + edge scatter ; out = act(agg + b)
template <int CI, int CO>
static inline void gcn_t(const float* Xin, const float* W, const float* B,
                         float* H, float* AGG, float* out,
                         const int* src, const int* dst, const float* dis,
                         int N, int E, int act, hipStream_t stream) {
  gemm_t<CI, CO>(Xin, W, nullptr, H, N, 0, stream);
  int nodeBlocks = (N + WPB - 1) / WPB;
  int edgeBlocks = (E + WPB - 1) / WPB;
  k_selfloop<<<nodeBlocks, TPB, 0, stream>>>(H, dis, AGG, N, CO);
  k_edge_agg<<<edgeBlocks, TPB, 0, stream>>>(H, src, dst, dis, AGG, E, CO);
  k_bias_act<<<nodeBlocks, TPB, 0, stream>>>(AGG, B, out, N, CO, act);
}

extern "C" void kernel_launch(void* const* d_in, const int* in_sizes, int n_in,
                              void* d_out, int out_size, void* d_ws, size_t ws_size,
                              hipStream_t stream) {
  const float* x  = (const float*)d_in[0];
  const int*   ei = (const int*)d_in[1];
  int N = in_sizes[0] / 3;
  int E = in_sizes[1] / 2;
  const int* src = ei;        // edge_index[0]
  const int* dst = ei + E;    // edge_index[1]

  const float* W[12];
  const float* B[12];
  for (int i = 0; i < 12; ++i) {
    W[i] = (const float*)d_in[2 + 2 * i];
    B[i] = (const float*)d_in[3 + 2 * i];
  }

  // Workspace (floats): deg[N] | dis[N] | bufH[N*160] | bufA[N*160]  (~129 MB)
  float* ws   = (float*)d_ws;
  float* deg  = ws;
  float* dis  = ws + (size_t)N;
  float* bufH = dis + (size_t)N;
  float* bufA = bufH + (size_t)N * 160;

  // Degrees (with self-loops) -> dis = deg^-1/2
  k_deg_init<<<(N + 255) / 256, 256, 0, stream>>>(deg, N);
  k_deg_edges<<<(E + 255) / 256, 256, 0, stream>>>(dst, deg, E);
  k_rsqrt<<<(N + 255) / 256, 256, 0, stream>>>(deg, dis, N);

  // Encoder GCN stack (relu)
  gcn_t<3, 160>(x,    W[0], B[0], bufH, bufA, bufA, src, dst, dis, N, E, 1, stream);
  gcn_t<160, 80>(bufA, W[1], B[1], bufH, bufA, bufA, src, dst, dis, N, E, 1, stream);
  gcn_t<80, 40>(bufA, W[2], B[2], bufH, bufA, bufA, src, dst, dis, N, E, 1, stream);
  gcn_t<40, 20>(bufA, W[3], B[3], bufH, bufA, bufA, src, dst, dis, N, E, 1, stream);

  // Dense latent layers (bias+act fused into GEMM epilogue)
  gemm_t<20, 10>(bufA, W[4], B[4], bufH, N, 1, stream);  // el1 relu
  gemm_t<10, 3>(bufH, W[5], B[5], bufA, N, 0, stream);   // el2 (latent)
  gemm_t<3, 10>(bufA, W[6], B[6], bufH, N, 1, stream);   // dl1 relu
  gemm_t<10, 20>(bufH, W[7], B[7], bufA, N, 1, stream);  // dl2 relu

  // Decoder GCN stack
  gcn_t<20, 40>(bufA, W[8],  B[8],  bufH, bufA, bufA, src, dst, dis, N, E, 1, stream);
  gcn_t<40, 80>(bufA, W[9],  B[9],  bufH, bufA, bufA, src, dst, dis, N, E, 1, stream);
  gcn_t<80, 160>(bufA, W[10], B[10], bufH, bufA, bufA, src, dst, dis, N, E, 1, stream);
  gcn_t<160, 3>(bufA, W[11], B[11], bufH, bufA, (float*)d_out, src, dst, dis, N, E, 2, stream);
}